// IVDetect_simplify_22857815949594
// MI455X (gfx1250) — compile-verified
//
#include <hip/hip_runtime.h>
#include <hip/hip_bf16.h>
#include <math.h>

// ---------------------------------------------------------------------------
// GCN 3-layer forward for MI455X (gfx1250, wave32).
//  - Dense GEMMs via V_WMMA_F32_16X16X4_F32 (fp32-exact vs reference),
//    2x M-blocked waves, guard-free inner loop over pre-padded transposed
//    weights (all fragment loads are global_load_b64).
//  - Intermediates padded to LD=208 (13 x 16): no N-dim guards anywhere.
//  - Edge gather/scatter: one wave per edge, native f32 global atomics;
//    both 83MB feature buffers fit MI455X's 192MB L2 -> L2-resident traffic.
// ---------------------------------------------------------------------------

#define DIN   100
#define DOUT  200
#define LD    208      // padded feature leading dimension (13*16)
#define NTN   (LD/16)  // 13 N-tiles

typedef float v2f __attribute__((ext_vector_type(2)));
typedef float v8f __attribute__((ext_vector_type(8)));

// ---- degree count: deg[dst] += 1 ------------------------------------------
__global__ void deg_kernel(const int* __restrict__ dst, float* __restrict__ deg,
                           int E) {
    int e = blockIdx.x * blockDim.x + threadIdx.x;
    if (e < E) unsafeAtomicAdd(&deg[dst[e]], 1.0f);
}

// ---- dinv[i] = rsqrt(deg[i] + 1) (in place) -------------------------------
__global__ void rsqrt_kernel(float* __restrict__ dinv, int n) {
    int i = blockIdx.x * blockDim.x + threadIdx.x;
    if (i < n) dinv[i] = rsqrtf(dinv[i] + 1.0f);
}

// ---- weight pad+transpose: Wt[208][K] = W[K][NC]^T, zero for col>=NC ------
__global__ void padw_kernel(const float* __restrict__ W, float* __restrict__ Wt,
                            int K, int NC) {
    int t = blockIdx.x * blockDim.x + threadIdx.x;   // over LD*K
    if (t >= LD * K) return;
    int c = t / K, k = t - c * K;
    Wt[t] = (c < NC) ? W[k * NC + c] : 0.0f;
}

// ---- bias pad: bp[208] = bias (zeros beyond NC, or all-zero) --------------
__global__ void padb_kernel(const float* __restrict__ b, float* __restrict__ bp,
                            int NC, int has) {
    int t = blockIdx.x * blockDim.x + threadIdx.x;
    if (t < LD) bp[t] = (has && t < NC) ? b[t] : 0.0f;
}

// ---- WMMA fp32 GEMM: C[M x 208] = A[M x K] * Wt^T (+ biasPad) -------------
// One wave per 32x16 tile (2 M-tiles, shared B fragment). K % 4 == 0.
// No guards: weights pre-padded/transposed, M % 32 == 0, LD % 16 == 0.
__global__ void gemm_wmma_kernel(const float* __restrict__ A, int lda,
                                 const float* __restrict__ Bt,      // [LD x K]
                                 const float* __restrict__ biasPad, // [LD]
                                 float* __restrict__ C,
                                 int M, int K) {
    const int lane = threadIdx.x & 31;
    const int wave = blockIdx.x * (blockDim.x >> 5) + (threadIdx.x >> 5);
    const int mt2  = wave / NTN;
    const int nt   = wave - mt2 * NTN;
    const int row0 = mt2 * 32;
    if (row0 >= M) return;                 // wave-uniform exit (EXEC stays full)

    const int lh  = lane & 15;
    const int hi  = lane >> 4;             // 0: K {k,k+1} | 1: K {k+2,k+3}
    const int col = nt * 16 + lh;

    const float cinit = biasPad[col];
    v8f acc0 = {cinit, cinit, cinit, cinit, cinit, cinit, cinit, cinit};
    v8f acc1 = acc0;

    const float* __restrict__ a0   = A  + (size_t)(row0 + lh) * lda;
    const float* __restrict__ a1   = a0 + (size_t)16 * lda;
    const float* __restrict__ bcol = Bt + (size_t)col * K;

#pragma unroll 2
    for (int k = 0; k < K; k += 4) {
        const int ka = k + hi * 2;
        v2f af0 = *(const v2f*)(a0 + ka);      // global_load_b64
        v2f af1 = *(const v2f*)(a1 + ka);      // global_load_b64
        v2f bf  = *(const v2f*)(bcol + ka);    // global_load_b64 (contiguous)
        acc0 = __builtin_amdgcn_wmma_f32_16x16x4_f32(
                   false, af0, false, bf, (short)0, acc0, false, false);
        acc1 = __builtin_amdgcn_wmma_f32_16x16x4_f32(
                   false, af1, false, bf, (short)0, acc1, false, false);
    }

    float* __restrict__ c0 = C + (size_t)(row0 + hi * 8) * LD + nt * 16 + lh;
#pragma unroll
    for (int v = 0; v < 8; ++v) c0[(size_t)v * LD]        = acc0[v];
#pragma unroll
    for (int v = 0; v < 8; ++v) c0[(size_t)(v + 16) * LD] = acc1[v];
}

// ---- edge scatter: hout[dst] += hin[src] * dinv[src]*dinv[dst] ------------
// One wave32 per edge; lanes stride the 200 feature dims (coalesced gather).
__global__ void scatter_kernel(const float* __restrict__ hin,
                               float* __restrict__ hout,
                               const int* __restrict__ src,
                               const int* __restrict__ dst,
                               const float* __restrict__ dinv, int E) {
    const int wave = blockIdx.x * (blockDim.x >> 5) + (threadIdx.x >> 5);
    const int lane = threadIdx.x & 31;
    if (wave >= E) return;
    const int   s    = src[wave];
    const int   t    = dst[wave];
    const float coef = dinv[s] * dinv[t];
    const float* __restrict__ hs = hin  + (size_t)s * LD;
    float*       __restrict__ ht = hout + (size_t)t * LD;
#pragma unroll
    for (int d = lane; d < DOUT; d += 32)
        unsafeAtomicAdd(&ht[d], hs[d] * coef);
}

// ---- combine: agg = act(agg + h*dinv^2 + b) -------------------------------
__global__ void combine_kernel(float* __restrict__ agg,
                               const float* __restrict__ h,
                               const float* __restrict__ dinv,
                               const float* __restrict__ bias,
                               int relu, int n) {
    long long idx = (long long)blockIdx.x * blockDim.x + threadIdx.x;
    if (idx >= (long long)n * LD) return;
    int i = (int)(idx / LD);
    int d = (int)(idx - (long long)i * LD);
    float b  = (d < DOUT) ? bias[d] : 0.0f;
    float di = dinv[i];
    float v  = agg[idx] + h[idx] * di * di + b;
    if (relu) v = fmaxf(v, 0.0f);
    agg[idx] = v;
}

// ---- segment-max pooling via monotone float->uint keys --------------------
__device__ __forceinline__ unsigned f2ord(float f) {
    unsigned u = __float_as_uint(f);
    return (u & 0x80000000u) ? ~u : (u | 0x80000000u);
}
__device__ __forceinline__ float ord2f(unsigned k) {
    unsigned u = (k & 0x80000000u) ? (k ^ 0x80000000u) : ~k;
    return __uint_as_float(u);
}

__global__ void pool_init_kernel(unsigned* __restrict__ keys, int n) {
    int i = blockIdx.x * blockDim.x + threadIdx.x;
    if (i < n) keys[i] = f2ord(-INFINITY);
}

__global__ void pool_kernel(const float* __restrict__ h3,
                            const int* __restrict__ batch,
                            unsigned* __restrict__ keys, int n) {
    long long t = (long long)blockIdx.x * blockDim.x + threadIdx.x;
    if (t >= (long long)n * DOUT) return;
    int i = (int)(t / DOUT);
    int d = (int)(t - (long long)i * DOUT);
    atomicMax(&keys[(long long)batch[i] * DOUT + d],
              f2ord(h3[(long long)i * LD + d]));
}

// ---- classifier + softmax (tiny: G x 200 x 2) -----------------------------
__global__ void cls_kernel(const unsigned* __restrict__ keys,
                           const float* __restrict__ Wcls,
                           const float* __restrict__ bcls,
                           float* __restrict__ out, int G) {
    int g = blockIdx.x * blockDim.x + threadIdx.x;
    if (g >= G) return;
    float l0 = bcls[0], l1 = bcls[1];
    for (int d = 0; d < DOUT; ++d) {
        float p = ord2f(keys[(long long)g * DOUT + d]);
        l0 += p * Wcls[d * 2 + 0];
        l1 += p * Wcls[d * 2 + 1];
    }
    float m  = fmaxf(l0, l1);
    float e0 = __expf(l0 - m), e1 = __expf(l1 - m);
    float s  = e0 + e1;
    out[g * 2 + 0] = e0 / s;
    out[g * 2 + 1] = e1 / s;
}

// ---------------------------------------------------------------------------
extern "C" void kernel_launch(void* const* d_in, const int* in_sizes, int n_in,
                              void* d_out, int out_size, void* d_ws, size_t ws_size,
                              hipStream_t stream) {
    const float* x    = (const float*)d_in[0];
    const int*   ei   = (const int*)  d_in[1];
    const int*   bat  = (const int*)  d_in[2];
    const float* W1   = (const float*)d_in[3];
    const float* b1   = (const float*)d_in[4];
    const float* Wc   = (const float*)d_in[5];
    const float* bc   = (const float*)d_in[6];
    const float* W2   = (const float*)d_in[7];
    const float* b2   = (const float*)d_in[8];
    const float* W3   = (const float*)d_in[9];
    const float* b3   = (const float*)d_in[10];
    const float* Wcls = (const float*)d_in[11];
    const float* bcls = (const float*)d_in[12];
    float* out = (float*)d_out;

    const int N = in_sizes[0] / DIN;     // 100000
    const int E = in_sizes[1] / 2;       // 1600000
    const int G = out_size / 2;          // 128
    const int* src = ei;
    const int* dst = ei + E;

    // workspace layout (floats)
    float*    ws    = (float*)d_ws;
    float*    dinv  = ws;                                   // N
    float*    bufA  = ws + 100352;                          // N*LD
    float*    bufB  = bufA + (size_t)N * LD;                // N*LD
    unsigned* keys  = (unsigned*)(bufB + (size_t)N * LD);   // G*DOUT
    float*    Wt    = (float*)(keys + (size_t)G * DOUT);    // LD*DOUT (max)
    float*    bp    = Wt + (size_t)LD * DOUT;               // LD

    const size_t featBytes = (size_t)N * LD * sizeof(float);
    const int waves = (N / 32) * NTN;                       // 3125*13
    const int gBlk  = (waves * 32 + 255) / 256;             // GEMM grid
    const int eBlk  = (E * 32 + 255) / 256;                 // scatter grid
    const int cBlk  = (int)(((long long)N * LD + 255) / 256);
    const int pBlk  = (int)(((long long)N * DOUT + 255) / 256);
    const int wBlk1 = (LD * DIN  + 255) / 256;              // pad W1
    const int wBlk2 = (LD * DOUT + 255) / 256;              // pad 200xK weights

    // --- degrees -> dinv ---
    hipMemsetAsync(dinv, 0, (size_t)N * sizeof(float), stream);
    deg_kernel  <<<(E + 255) / 256, 256, 0, stream>>>(dst, dinv, E);
    rsqrt_kernel<<<(N + 255) / 256, 256, 0, stream>>>(dinv, N);

    // --- conv1: t1 = x@W1 ; agg ; h1 = relu(agg + t1*dinv^2 + b1) ---
    padw_kernel<<<wBlk1, 256, 0, stream>>>(W1, Wt, DIN, DOUT);
    padb_kernel<<<1, 256, 0, stream>>>(nullptr, bp, DOUT, 0);
    gemm_wmma_kernel<<<gBlk, 256, 0, stream>>>(x, DIN, Wt, bp, bufA, N, DIN);
    hipMemsetAsync(bufB, 0, featBytes, stream);
    scatter_kernel<<<eBlk, 256, 0, stream>>>(bufA, bufB, src, dst, dinv, E);
    combine_kernel<<<cBlk, 256, 0, stream>>>(bufB, bufA, dinv, b1, 1, N); // h1=bufB

    // --- h = h1@Wc + bc ---
    padw_kernel<<<wBlk2, 256, 0, stream>>>(Wc, Wt, DOUT, DOUT);
    padb_kernel<<<1, 256, 0, stream>>>(bc, bp, DOUT, 1);
    gemm_wmma_kernel<<<gBlk, 256, 0, stream>>>(bufB, LD, Wt, bp, bufA, N, DOUT);

    // --- conv2 ---
    padw_kernel<<<wBlk2, 256, 0, stream>>>(W2, Wt, DOUT, DOUT);
    padb_kernel<<<1, 256, 0, stream>>>(nullptr, bp, DOUT, 0);
    gemm_wmma_kernel<<<gBlk, 256, 0, stream>>>(bufA, LD, Wt, bp, bufB, N, DOUT);
    hipMemsetAsync(bufA, 0, featBytes, stream);
    scatter_kernel<<<eBlk, 256, 0, stream>>>(bufB, bufA, src, dst, dinv, E);
    combine_kernel<<<cBlk, 256, 0, stream>>>(bufA, bufB, dinv, b2, 1, N); // h2=bufA

    // --- conv3 (no relu) ---
    padw_kernel<<<wBlk2, 256, 0, stream>>>(W3, Wt, DOUT, DOUT);
    gemm_wmma_kernel<<<gBlk, 256, 0, stream>>>(bufA, LD, Wt, bp, bufB, N, DOUT);
    hipMemsetAsync(bufA, 0, featBytes, stream);
    scatter_kernel<<<eBlk, 256, 0, stream>>>(bufB, bufA, src, dst, dinv, E);
    combine_kernel<<<cBlk, 256, 0, stream>>>(bufA, bufB, dinv, b3, 0, N); // h3=bufA

    // --- global max pool + classifier + softmax ---
    pool_init_kernel<<<(G * DOUT + 255) / 256, 256, 0, stream>>>(keys, G * DOUT);
    pool_kernel<<<pBlk, 256, 0, stream>>>(bufA, bat, keys, N);
    cls_kernel<<<(G + 127) / 128, 128, 0, stream>>>(keys, Wcls, bcls, out, G);
}